// DNANet_4037269258360
// MI455X (gfx1250) — compile-verified
//
#include <hip/hip_runtime.h>
#include <math.h>

#define N_NODES 10000
#define FIN     256
#define HDIM    128
#define COUT    40
#define NHEADS  8
#define DHEAD   16
#define NEDGE   160000
#define ETOT    (NEDGE + N_NODES)          // edges + self loops = 170000
#define NH      (N_NODES * HDIM)           // 1,280,000 floats per node-feature slice
#define ATT_SCALE 0.25f                    // 1/sqrt(16)

typedef __attribute__((ext_vector_type(2))) float v2f;
typedef __attribute__((ext_vector_type(8))) float v8f;

// ---------------- degree / norm ----------------
__global__ void deg_kernel(const int* __restrict__ ecol, float* __restrict__ deg) {
  int e = blockIdx.x * blockDim.x + threadIdx.x;
  if (e >= ETOT) return;
  int c = (e < NEDGE) ? ecol[e] : (e - NEDGE);
  __hip_atomic_fetch_add(&deg[c], 1.0f, __ATOMIC_RELAXED, __HIP_MEMORY_SCOPE_AGENT);
}

__global__ void rsqrt_kernel(float* __restrict__ d) {
  int i = blockIdx.x * blockDim.x + threadIdx.x;
  if (i < N_NODES) d[i] = rsqrtf(fmaxf(d[i], 1e-12f));   // deg >= 1 (self loops)
}

// ---------------- fp32 WMMA GEMM: C = relu?(A[M,K] @ B[K,Ncols] + bias) ----------------
// One wave per 16x16 tile of C.  Requires: M % 16 == 0, Ncols % 16 == 0, K % 8 == 0.
// Branchless inner loop: A pair as one b64 load, two b32 B loads, raw-pointer bumps.
__global__ void wmma_gemm_f32(const float* __restrict__ A, const float* __restrict__ B,
                              const float* __restrict__ bias, float* __restrict__ C,
                              int Ncols, int K, int relu) {
  const int lane = threadIdx.x;          // 0..31, wave32
  const int tm   = blockIdx.y * 16;
  const int tn   = blockIdx.x * 16;
  const int half = lane >> 4;            // 0: K pair {0,1}; 1: K pair {2,3}
  const int l16  = lane & 15;
  const int arow = tm + l16;             // A fragment: both lane halves map M=0..15
  const int bcol = tn + l16;             // B/C fragment: N = lane % 16

  const float* __restrict__ Ap = A + (size_t)arow * K + half * 2;
  const float* __restrict__ Bp = B + (size_t)(half * 2) * Ncols + bcol;
  const size_t bstep = (size_t)Ncols;

  v8f acc = {};
  for (int k0 = 0; k0 < K; k0 += 8) {
    v2f a0 = *reinterpret_cast<const v2f*>(Ap);
    v2f b0; b0.x = Bp[0]; b0.y = Bp[bstep];
    v2f a1 = *reinterpret_cast<const v2f*>(Ap + 4);
    v2f b1; b1.x = Bp[4 * bstep]; b1.y = Bp[5 * bstep];
    acc = __builtin_amdgcn_wmma_f32_16x16x4_f32(false, a0, false, b0,
                                                (short)0, acc, false, false);
    acc = __builtin_amdgcn_wmma_f32_16x16x4_f32(false, a1, false, b1,
                                                (short)0, acc, false, false);
    Ap += 8;
    Bp += 8 * bstep;
  }

  const float bb = bias ? bias[bcol] : 0.0f;
  float* __restrict__ Cp = C + (size_t)(tm + half * 8) * Ncols + bcol;
#pragma unroll
  for (int r = 0; r < 8; ++r) {          // C VGPR r: M = r (lanes 0-15) / r+8 (lanes 16-31)
    float val = acc[r] + bb;
    if (relu) val = fmaxf(val, 0.0f);
    Cp[(size_t)r * Ncols] = val;
  }
}

// ---------------- per-(edge,head) attention + scatter ----------------
__global__ void edge_attn_kernel(const int* __restrict__ erow, const int* __restrict__ ecol,
                                 const float* __restrict__ dis, const float* __restrict__ q,
                                 const float* __restrict__ k, const float* __restrict__ v,
                                 float* __restrict__ out, int t) {
  int idx = blockIdx.x * blockDim.x + threadIdx.x;
  if (idx >= ETOT * NHEADS) return;
  const int e = idx >> 3;
  const int h = idx & 7;
  int r, c;
  if (e < NEDGE) { r = erow[e]; c = ecol[e]; } else { r = e - NEDGE; c = r; }
  const float nrm = dis[r] * dis[c];

  float qv[DHEAD];
  {
    const float4* qp = reinterpret_cast<const float4*>(q + (size_t)c * HDIM + h * DHEAD);
#pragma unroll
    for (int i = 0; i < 4; ++i) reinterpret_cast<float4*>(qv)[i] = qp[i];
  }

  float score[3];
  for (int j = 0; j < t; ++j) {
    const float4* kp = reinterpret_cast<const float4*>(k + (size_t)j * NH + (size_t)r * HDIM + h * DHEAD);
    float kv[DHEAD];
#pragma unroll
    for (int i = 0; i < 4; ++i) reinterpret_cast<float4*>(kv)[i] = kp[i];
    float s = 0.0f;
#pragma unroll
    for (int i = 0; i < DHEAD; ++i) s += qv[i] * kv[i];
    score[j] = s * ATT_SCALE;
  }

  // restricted softmax: implicit zero logit
  float m = 0.0f;
  for (int j = 0; j < t; ++j) m = fmaxf(m, score[j]);
  float denom = expf(-m);
  float w[3];
  for (int j = 0; j < t; ++j) { w[j] = expf(score[j] - m); denom += w[j]; }
  const float scl = nrm / denom;

  float acc[DHEAD];
#pragma unroll
  for (int i = 0; i < DHEAD; ++i) acc[i] = 0.0f;
  for (int j = 0; j < t; ++j) {
    const float aw = w[j] * scl;
    const float4* vp = reinterpret_cast<const float4*>(v + (size_t)j * NH + (size_t)r * HDIM + h * DHEAD);
    float vv[DHEAD];
#pragma unroll
    for (int i = 0; i < 4; ++i) reinterpret_cast<float4*>(vv)[i] = vp[i];
#pragma unroll
    for (int i = 0; i < DHEAD; ++i) acc[i] += aw * vv[i];
  }

  float* op = out + (size_t)c * HDIM + h * DHEAD;
#pragma unroll
  for (int i = 0; i < DHEAD; ++i)
    __hip_atomic_fetch_add(&op[i], acc[i], __ATOMIC_RELAXED, __HIP_MEMORY_SCOPE_AGENT);
}

__global__ void relu_kernel(const float* __restrict__ in, float* __restrict__ outp, int n) {
  int i = blockIdx.x * blockDim.x + threadIdx.x;
  if (i < n) outp[i] = fmaxf(in[i], 0.0f);
}

// ---------------- lin2 + log_softmax, one 64-thread block per node ----------------
__global__ void classify_kernel(const float* __restrict__ x, const float* __restrict__ w,
                                const float* __restrict__ b, float* __restrict__ out) {
  __shared__ float xs[HDIM];
  __shared__ float lg[COUT];
  const int node = blockIdx.x;
  const int tid = threadIdx.x;          // 64 threads
  for (int i = tid; i < HDIM; i += 64) xs[i] = x[(size_t)node * HDIM + i];
  __syncthreads();
  if (tid < COUT) {
    float s = b[tid];
    for (int i = 0; i < HDIM; ++i) s += xs[i] * w[i * COUT + tid];
    lg[tid] = s;
  }
  __syncthreads();
  if (tid < COUT) {
    float m = -INFINITY;
    for (int i = 0; i < COUT; ++i) m = fmaxf(m, lg[i]);
    float se = 0.0f;
    for (int i = 0; i < COUT; ++i) se += expf(lg[i] - m);
    out[(size_t)node * COUT + tid] = lg[tid] - m - logf(se);
  }
}

extern "C" void kernel_launch(void* const* d_in, const int* in_sizes, int n_in,
                              void* d_out, int out_size, void* d_ws, size_t ws_size,
                              hipStream_t stream) {
  (void)in_sizes; (void)n_in; (void)out_size; (void)ws_size;
  const int*   edge    = (const int*)d_in[0];
  const int*   erow    = edge;               // edge_index[0]
  const int*   ecol    = edge + NEDGE;       // edge_index[1]
  const float* x_param = (const float*)d_in[1];
  const float* lin1_w  = (const float*)d_in[2];
  const float* lin1_b  = (const float*)d_in[3];
  const float* wq      = (const float*)d_in[4];
  const float* wk      = (const float*)d_in[5];
  const float* wv      = (const float*)d_in[6];
  const float* bq      = (const float*)d_in[7];
  const float* bk      = (const float*)d_in[8];
  const float* bv      = (const float*)d_in[9];
  const float* lin2_w  = (const float*)d_in[10];
  const float* lin2_b  = (const float*)d_in[11];
  float* outp = (float*)d_out;

  // scratch layout (floats): dis[10016] | xs[4*NH] | q[NH] | k[3*NH] | v[3*NH] | ob[NH]  (~62 MB)
  float* ws  = (float*)d_ws;
  float* dis = ws;
  float* xs  = ws + 10016;
  float* qb  = xs + 4 * (size_t)NH;
  float* kb  = qb + (size_t)NH;
  float* vb  = kb + 3 * (size_t)NH;
  float* ob  = vb + 3 * (size_t)NH;

  const int thr = 256;

  // 1) gcn_norm
  hipMemsetAsync(dis, 0, N_NODES * sizeof(float), stream);
  deg_kernel<<<(ETOT + thr - 1) / thr, thr, 0, stream>>>(ecol, dis);
  rsqrt_kernel<<<(N_NODES + thr - 1) / thr, thr, 0, stream>>>(dis);

  // 2) x0 = relu(x_param @ lin1_w + lin1_b)
  dim3 gH(HDIM / 16, N_NODES / 16);   // (8, 625), one wave per tile
  wmma_gemm_f32<<<gH, 32, 0, stream>>>(x_param, lin1_w, lin1_b, xs, HDIM, FIN, 1);

  // 3) transformer layers
  for (int l = 0; l < 3; ++l) {
    const int t = l + 1;
    wmma_gemm_f32<<<gH, 32, 0, stream>>>(xs + (size_t)l * NH, wq + (size_t)l * HDIM * HDIM,
                                         bq + l * HDIM, qb, HDIM, HDIM, 0);
    for (int j = 0; j < t; ++j) {
      wmma_gemm_f32<<<gH, 32, 0, stream>>>(xs + (size_t)j * NH, wk + (size_t)l * HDIM * HDIM,
                                           bk + l * HDIM, kb + (size_t)j * NH, HDIM, HDIM, 0);
      wmma_gemm_f32<<<gH, 32, 0, stream>>>(xs + (size_t)j * NH, wv + (size_t)l * HDIM * HDIM,
                                           bv + l * HDIM, vb + (size_t)j * NH, HDIM, HDIM, 0);
    }
    hipMemsetAsync(ob, 0, (size_t)NH * sizeof(float), stream);
    const int nwork = ETOT * NHEADS;    // 1,360,000
    edge_attn_kernel<<<(nwork + thr - 1) / thr, thr, 0, stream>>>(erow, ecol, dis, qb, kb, vb, ob, t);
    relu_kernel<<<(NH + thr - 1) / thr, thr, 0, stream>>>(ob, xs + (size_t)(l + 1) * NH, NH);
  }

  // 4) logits + log_softmax from x_all[:, -1] == x3
  classify_kernel<<<N_NODES, 64, 0, stream>>>(xs + 3 * (size_t)NH, lin2_w, lin2_b, outp);
}